// GCN_73830487818944
// MI455X (gfx1250) — compile-verified
//
#include <hip/hip_runtime.h>
#include <math.h>

// Problem constants (fixed by the reference).
#define NN 100000   // nodes
#define EE 1000000  // edges
#define FF 128      // input features
#define DD 64       // embedding size
#define GG 256      // graphs

typedef float v2f __attribute__((ext_vector_type(2)));
typedef float v8f __attribute__((ext_vector_type(8)));

// ---------------- degree / symmetric norm ----------------
__global__ void k_zero_deg(float* __restrict__ deg) {
  int i = blockIdx.x * blockDim.x + threadIdx.x;
  if (i < NN) deg[i] = 0.0f;
}

__global__ void k_degree(const int* __restrict__ dst, float* __restrict__ deg) {
  int e = blockIdx.x * blockDim.x + threadIdx.x;
  if (e < EE) atomicAdd(&deg[dst[e]], 1.0f);
}

__global__ void k_dinv(float* __restrict__ deg) {
  int i = blockIdx.x * blockDim.x + threadIdx.x;
  if (i < NN) deg[i] = rsqrtf(deg[i] + 1.0f);   // dinv, in place
}

// ---------------- fp32 WMMA GEMM: C[NN x 64] = A[NN x K] * B[K x 64] ----------------
// One wave computes one 16x16 tile using V_WMMA_F32_16X16X4_F32.
// A-frag (16x4): lanes 0-15 -> M=lane, K={2h, 2h+1} with h = lane>>4 selecting K-pairs.
// B-frag (4x16): VGPR j, lanes 0-15 -> K=j, lanes 16-31 -> K=j+2, N=lane&15 within tile.
// C/D (16x16):   VGPR j, lanes 0-15 -> M=j, lanes 16-31 -> M=j+8, N=lane&15.
template <int K>
__global__ void k_gemm_wmma(const float* __restrict__ A, const float* __restrict__ B,
                            float* __restrict__ C) {
  int gid  = blockIdx.x * blockDim.x + threadIdx.x;
  int wave = gid >> 5;
  int lane = gid & 31;
  int tm = wave >> 2;          // 4 column tiles (DD/16)
  int tn = wave & 3;
  if (tm * 16 >= NN) return;   // never taken with exact grid; keeps EXEC uniform

  int half = lane >> 4;        // 0: K pair {0,1}; 1: K pair {2,3}
  int m    = lane & 15;
  const float* arow = A + (size_t)(tm * 16 + m) * K;
  int col = tn * 16 + m;

  v8f acc = {};
#pragma unroll
  for (int kb = 0; kb < K; kb += 4) {
    int k0 = kb + half * 2;
    v2f a; a[0] = arow[k0];          a[1] = arow[k0 + 1];
    v2f b; b[0] = B[k0 * DD + col];  b[1] = B[(k0 + 1) * DD + col];
    acc = __builtin_amdgcn_wmma_f32_16x16x4_f32(
        /*neg_a=*/false, a, /*neg_b=*/false, b,
        /*c_mod=*/(short)0, acc, /*reuse_a=*/false, /*reuse_b=*/false);
  }

  int r0 = tm * 16 + half * 8;
#pragma unroll
  for (int j = 0; j < 8; ++j)
    C[(size_t)(r0 + j) * DD + col] = acc[j];
}

// ---------------- aggregation ----------------
// agg = self_norm * hw  (self_norm = dinv^2)
__global__ void k_init_agg(const float* __restrict__ hw, const float* __restrict__ dinv,
                           float* __restrict__ agg) {
  int i = blockIdx.x * blockDim.x + threadIdx.x;
  if (i < NN * DD) {
    float dv = dinv[i >> 6];
    agg[i] = dv * dv * hw[i];
  }
}

// agg[dst] += dinv[src]*dinv[dst] * hw[src]   (one wave per edge, 2 floats/lane)
__global__ void k_scatter(const int* __restrict__ src, const int* __restrict__ dst,
                          const float* __restrict__ dinv, const float* __restrict__ hw,
                          float* __restrict__ agg) {
  int gid  = blockIdx.x * blockDim.x + threadIdx.x;
  int e    = gid >> 5;
  int lane = gid & 31;
  if (e >= EE) return;
  int s = src[e], d = dst[e];
  float w = dinv[s] * dinv[d];
  const float2* hp = (const float2*)(hw + (size_t)s * DD);
  float2 v = hp[lane];
  float* ap = agg + (size_t)d * DD + lane * 2;
  atomicAdd(ap,     v.x * w);
  atomicAdd(ap + 1, v.y * w);
}

// h = tanh(agg + b)  (in place)
__global__ void k_finish(const float* __restrict__ b, float* __restrict__ agg) {
  int i = blockIdx.x * blockDim.x + threadIdx.x;
  if (i < NN * DD) agg[i] = tanhf(agg[i] + b[i & 63]);
}

// ---------------- pooling ----------------
__device__ __forceinline__ unsigned fflip(float f) {
  unsigned u = __float_as_uint(f);
  return u ^ ((unsigned)((int)u >> 31) | 0x80000000u);   // monotone float->uint
}
__device__ __forceinline__ float funflip(unsigned u) {
  unsigned m = (u >> 31) ? 0x80000000u : 0xFFFFFFFFu;
  return __uint_as_float(u ^ m);
}

__global__ void k_pool_init(unsigned* __restrict__ gmp, float* __restrict__ gsum,
                            float* __restrict__ counts) {
  int i = blockIdx.x * blockDim.x + threadIdx.x;
  if (i < GG * DD) { gmp[i] = 0x007FFFFFu /* fflip(-inf) */; gsum[i] = 0.0f; }
  if (i < GG) counts[i] = 0.0f;
}

__global__ void k_pool(const float* __restrict__ h, const int* __restrict__ bidx,
                       unsigned* __restrict__ gmp, float* __restrict__ gsum,
                       float* __restrict__ counts) {
  int i = blockIdx.x * blockDim.x + threadIdx.x;
  if (i >= NN * DD) return;
  int n = i >> 6, dd = i & 63;
  int g = bidx[n];
  float v = h[i];
  atomicMax(&gmp[g * DD + dd], fflip(v));
  atomicAdd(&gsum[g * DD + dd], v);
  if (dd == 0) atomicAdd(&counts[g], 1.0f);
}

// hidden = [gmp | gsum/count]; out = hidden @ Wout + bout
__global__ void k_final(const unsigned* __restrict__ gmp, const float* __restrict__ gsum,
                        const float* __restrict__ counts, const float* __restrict__ Wout,
                        const float* __restrict__ bout, float* __restrict__ out,
                        float* __restrict__ hidden) {
  __shared__ float red[2 * DD];
  int g = blockIdx.x, t = threadIdx.x;
  float cnt = fmaxf(counts[g], 1.0f);
  float hv = (t < DD) ? funflip(gmp[g * DD + t]) : (gsum[g * DD + (t - DD)] / cnt);
  hidden[(size_t)g * 2 * DD + t] = hv;
  red[t] = hv * Wout[t];
  __syncthreads();
  for (int s = DD; s > 0; s >>= 1) {
    if (t < s) red[t] += red[t + s];
    __syncthreads();
  }
  if (t == 0) out[g] = red[0] + bout[0];
}

// ---------------- host launch ----------------
extern "C" void kernel_launch(void* const* d_in, const int* in_sizes, int n_in,
                              void* d_out, int out_size, void* d_ws, size_t ws_size,
                              hipStream_t stream) {
  const float* x    = (const float*)d_in[0];
  const int*   ei   = (const int*)d_in[1];
  const int*   bidx = (const int*)d_in[2];
  const float* W0   = (const float*)d_in[3];
  const float* b0   = (const float*)d_in[4];
  const float* W1   = (const float*)d_in[5];
  const float* b1   = (const float*)d_in[6];
  const float* W2   = (const float*)d_in[7];
  const float* b2   = (const float*)d_in[8];
  const float* W3   = (const float*)d_in[9];
  const float* b3   = (const float*)d_in[10];
  const float* Wout = (const float*)d_in[11];
  const float* bout = (const float*)d_in[12];

  const int* src = ei;        // edge_index[0, :]
  const int* dst = ei + EE;   // edge_index[1, :]

  // workspace layout
  char*  ws   = (char*)d_ws;
  size_t off  = 0;
  float* dinv = (float*)(ws + off);  off += (((size_t)NN * 4 + 255) & ~(size_t)255);
  float* bufP = (float*)(ws + off);  off += (size_t)NN * DD * 4;   // agg / h
  float* bufQ = (float*)(ws + off);  off += (size_t)NN * DD * 4;   // hw
  unsigned* gmp = (unsigned*)(ws + off); off += (size_t)GG * DD * 4;
  float* gsum   = (float*)(ws + off);    off += (size_t)GG * DD * 4;
  float* counts = (float*)(ws + off);    off += (size_t)GG * 4;
  (void)ws_size; (void)n_in; (void)in_sizes; (void)out_size;

  float* outp   = (float*)d_out;   // [G]
  float* hidden = outp + GG;       // [G, 2D]

  const int T = 256;

  // normalization
  k_zero_deg<<<(NN + T - 1) / T, T, 0, stream>>>(dinv);
  k_degree  <<<(EE + T - 1) / T, T, 0, stream>>>(dst, dinv);
  k_dinv    <<<(NN + T - 1) / T, T, 0, stream>>>(dinv);

  const int gemm_blocks = (NN / 16) * 4 / 8;     // 25000 waves / 8 waves per block
  const int nd_blocks   = (NN * DD + T - 1) / T;
  const int sc_blocks   = (int)(((size_t)EE * 32 + T - 1) / T);

  const float* Wl[4] = {W0, W1, W2, W3};
  const float* bl[4] = {b0, b1, b2, b3};
  for (int l = 0; l < 4; ++l) {
    if (l == 0) k_gemm_wmma<FF><<<gemm_blocks, T, 0, stream>>>(x,    Wl[l], bufQ);
    else        k_gemm_wmma<DD><<<gemm_blocks, T, 0, stream>>>(bufP, Wl[l], bufQ);
    k_init_agg<<<nd_blocks, T, 0, stream>>>(bufQ, dinv, bufP);
    k_scatter <<<sc_blocks, T, 0, stream>>>(src, dst, dinv, bufQ, bufP);
    k_finish  <<<nd_blocks, T, 0, stream>>>(bl[l], bufP);
  }

  // pooling + readout
  k_pool_init<<<(GG * DD + T - 1) / T, T, 0, stream>>>(gmp, gsum, counts);
  k_pool     <<<nd_blocks, T, 0, stream>>>(bufP, bidx, gmp, gsum, counts);
  k_final    <<<GG, 2 * DD, 0, stream>>>(gmp, gsum, counts, Wout, bout, outp, hidden);
}